// TensorProduct_7232724927057
// MI455X (gfx1250) — compile-verified
//
#include <hip/hip_runtime.h>
#include <stdint.h>

typedef float v2f __attribute__((ext_vector_type(2)));
typedef float v8f __attribute__((ext_vector_type(8)));

#define IRR     9
#define CH      64
#define MAXNNZ  160
#define MAXPATH 16
#define WAVES   4
#define TPB     (WAVES * 32)
#define BPW     8              // batch elements per wave (double-buffered)

// out[b, mo, c] = sum_k A[mo,k] * B[b,k,c]
//   A[mo,k] = (idx_out[k]==mo) ? cg[k] : 0          (static scatter matrix)
//   B[k,c]  = in1[b,idx1[k],c]*in2[b,idx2[k],c]*w[path[k],c]
// via V_WMMA_F32_16X16X4_F32, K-chunks of 4, four 16-channel N tiles.
// Inputs are double-buffer streamed into LDS with global_load_async_to_lds.
__global__ __launch_bounds__(TPB) void tp_cg_wmma_kernel(
    const float* __restrict__ in1, const float* __restrict__ in2,
    const float* __restrict__ w,  const float* __restrict__ cg,
    const int* __restrict__ i1,   const int* __restrict__ i2,
    const int* __restrict__ io,   const int* __restrict__ pp,
    float* __restrict__ out, int nnz, int npaths, int nbatch)
{
    __shared__ int4  sMeta[MAXNNZ];           // {i1*CH, i2*CH, path*CH, idx_out}
    __shared__ float sCG[MAXNNZ];
    __shared__ float sW[MAXPATH * CH];
    __shared__ float sIn1[WAVES][2][IRR * CH];
    __shared__ float sIn2[WAVES][2][IRR * CH];

    const int tid  = threadIdx.x;
    const int lane = tid & 31;
    const int wave = tid >> 5;

    if (nnz > MAXNNZ) nnz = MAXNNZ;
    int nnzp = (nnz + 3) & ~3;
    if (nnzp < 4) nnzp = 4;     // padding entries are always safe (A contributes 0)

    // ---- cooperative stage of COO tables + weights into LDS ----
    for (int k = tid; k < nnzp; k += TPB) {
        if (k < nnz) {
            sMeta[k] = make_int4(i1[k] * CH, i2[k] * CH, pp[k] * CH, io[k]);
            sCG[k]   = cg[k];
        } else {            // K padding: idx_out=255 never matches -> A = 0
            sMeta[k] = make_int4(0, 0, 0, 255);
            sCG[k]   = 0.0f;
        }
    }
    for (int k = tid; k < npaths * CH && k < MAXPATH * CH; k += TPB) sW[k] = w[k];
    __syncthreads();

    const int bBase = (blockIdx.x * WAVES + wave) * BPW;
    if (bBase >= nbatch) return;  // wave-uniform

    const uint32_t l1base = (uint32_t)(uintptr_t)(&sIn1[wave][0][0]);
    const uint32_t l2base = (uint32_t)(uintptr_t)(&sIn2[wave][0][0]);
    const int clampB = nbatch - 1;

    // ---- async stage of one (in1,in2) tile into LDS buffer `buf` ----
    auto stage = [&](int buf, int b) {
        const uint64_t g1 = (uint64_t)(uintptr_t)(in1 + (size_t)b * (IRR * CH));
        const uint64_t g2 = (uint64_t)(uintptr_t)(in2 + (size_t)b * (IRR * CH));
        const uint32_t l1 = l1base + (uint32_t)buf * (IRR * CH * 4);
        const uint32_t l2 = l2base + (uint32_t)buf * (IRR * CH * 4);
        // 9*64 floats = 288 x 8B = 9 iterations x 32 lanes, b64 each => 18 loads
        #pragma unroll
        for (int it = 0; it < 9; ++it) {
            const uint32_t off = (uint32_t)((it * 32 + lane) * 8);
            uint32_t d1 = l1 + off, d2 = l2 + off;
            uint64_t a1 = g1 + off, a2 = g2 + off;
            asm volatile("global_load_async_to_lds_b64 %0, %1, off"
                         :: "v"(d1), "v"(a1) : "memory");
            asm volatile("global_load_async_to_lds_b64 %0, %1, off"
                         :: "v"(d2), "v"(a2) : "memory");
        }
    };

    stage(0, bBase);   // prologue

    const int m     = lane & 15;            // output row this lane owns in A
    const int cl    = lane & 15;            // channel-within-tile for B/C/D
    const int khalf = (lane >> 4) * 2;      // K sub-offset per lane half

    for (int i = 0; i < BPW; ++i) {
        const int b   = bBase + i;
        if (b >= nbatch) break;             // wave-uniform

        // issue next tile into the other buffer, then release current tile
        if (i + 1 < BPW && b + 1 <= clampB) {
            stage((i + 1) & 1, b + 1);
            asm volatile("s_wait_asynccnt 0x12" ::: "memory");  // <=18: cur done
        } else {
            asm volatile("s_wait_asynccnt 0x0" ::: "memory");
        }

        const float* __restrict__ s1 = &sIn1[wave][i & 1][0];
        const float* __restrict__ s2 = &sIn2[wave][i & 1][0];

        v8f acc0 = {}, acc1 = {}, acc2 = {}, acc3 = {};

        int kc = 0;
        do {
            const int k0 = kc + khalf;
            const int k1 = k0 + 1;

            // hoisted unconditional LDS loads; A built with cndmask (no branch)
            const int4  mt0 = sMeta[k0];
            const int4  mt1 = sMeta[k1];
            const float c0  = sCG[k0];
            const float c1  = sCG[k1];

            v2f A;
            A.x = (mt0.w == m) ? c0 : 0.0f;
            A.y = (mt1.w == m) ? c1 : 0.0f;

            {
                const int c = 0 * 16 + cl;
                v2f B;
                B.x = s1[mt0.x + c] * s2[mt0.y + c] * sW[mt0.z + c];
                B.y = s1[mt1.x + c] * s2[mt1.y + c] * sW[mt1.z + c];
                acc0 = __builtin_amdgcn_wmma_f32_16x16x4_f32(false, A, false, B,
                                                             (short)0, acc0, false, false);
            }
            {
                const int c = 1 * 16 + cl;
                v2f B;
                B.x = s1[mt0.x + c] * s2[mt0.y + c] * sW[mt0.z + c];
                B.y = s1[mt1.x + c] * s2[mt1.y + c] * sW[mt1.z + c];
                acc1 = __builtin_amdgcn_wmma_f32_16x16x4_f32(false, A, false, B,
                                                             (short)0, acc1, false, false);
            }
            {
                const int c = 2 * 16 + cl;
                v2f B;
                B.x = s1[mt0.x + c] * s2[mt0.y + c] * sW[mt0.z + c];
                B.y = s1[mt1.x + c] * s2[mt1.y + c] * sW[mt1.z + c];
                acc2 = __builtin_amdgcn_wmma_f32_16x16x4_f32(false, A, false, B,
                                                             (short)0, acc2, false, false);
            }
            {
                const int c = 3 * 16 + cl;
                v2f B;
                B.x = s1[mt0.x + c] * s2[mt0.y + c] * sW[mt0.z + c];
                B.y = s1[mt1.x + c] * s2[mt1.y + c] * sW[mt1.z + c];
                acc3 = __builtin_amdgcn_wmma_f32_16x16x4_f32(false, A, false, B,
                                                             (short)0, acc3, false, false);
            }
            kc += 4;
        } while (kc < nnzp);

        // C/D layout: VGPR r -> lanes0-15 M=r, lanes16-31 M=r+8.
        // Rows 0-7 (lanes 0-15) are all valid; lanes 16-31 only row 8 (r==0).
        float* __restrict__ ob = out + (size_t)b * (IRR * CH) + cl;
        if (lane < 16) {
            #pragma unroll
            for (int r = 0; r < 8; ++r) {
                __builtin_nontemporal_store(acc0[r], ob + r * CH +  0);
                __builtin_nontemporal_store(acc1[r], ob + r * CH + 16);
                __builtin_nontemporal_store(acc2[r], ob + r * CH + 32);
                __builtin_nontemporal_store(acc3[r], ob + r * CH + 48);
            }
        } else {
            __builtin_nontemporal_store(acc0[0], ob + 8 * CH +  0);
            __builtin_nontemporal_store(acc1[0], ob + 8 * CH + 16);
            __builtin_nontemporal_store(acc2[0], ob + 8 * CH + 32);
            __builtin_nontemporal_store(acc3[0], ob + 8 * CH + 48);
        }
    }
}

extern "C" void kernel_launch(void* const* d_in, const int* in_sizes, int n_in,
                              void* d_out, int out_size, void* d_ws, size_t ws_size,
                              hipStream_t stream) {
    const float* in1 = (const float*)d_in[0];
    const float* in2 = (const float*)d_in[1];
    const float* w   = (const float*)d_in[2];
    const float* cg  = (const float*)d_in[3];
    const int*   i1  = (const int*)d_in[4];
    const int*   i2  = (const int*)d_in[5];
    const int*   io  = (const int*)d_in[6];
    const int*   pp  = (const int*)d_in[7];
    float* out = (float*)d_out;

    const int nnz    = in_sizes[3];
    const int npaths = in_sizes[2] / CH;
    const int nbatch = in_sizes[0] / (IRR * CH);
    const int perBlk = WAVES * BPW;
    const int grid   = (nbatch + perBlk - 1) / perBlk;

    tp_cg_wmma_kernel<<<grid, TPB, 0, stream>>>(in1, in2, w, cg, i1, i2, io, pp,
                                                out, nnz, npaths, nbatch);
}